// GNBlock_377957122655
// MI455X (gfx1250) — compile-verified
//
#include <hip/hip_runtime.h>
#include <hip/hip_bf16.h>

// ---------------------------------------------------------------------------
// GIN conv + MLP(128->512->256) + LeakyReLU + BatchNorm, fp32 end-to-end.
// GEMMs use CDNA5 V_WMMA_F32_16X16X4_F32 (fp32-exact matrix pipe).
// Wave tile = 16x64 (4 accumulators) to reuse each A fragment 4x and cut
// L2 read traffic 4x vs the 16x16-per-wave version.
// ---------------------------------------------------------------------------

typedef __attribute__((ext_vector_type(2))) float v2f;
typedef __attribute__((ext_vector_type(8))) float v8f;

#define N_NODES 50000
#define N_EDGES 800000
#define D_IN    128
#define D_HID   512
#define D_OUT   256
#define BN_EPS  1e-5
#define SLOPE   0.01f

// ---- h0 = x -----------------------------------------------------------------
__global__ void gn_copy_x(const float* __restrict__ x, float* __restrict__ h0, int n4) {
    int i = blockIdx.x * blockDim.x + threadIdx.x;
    if (i < n4) ((float4*)h0)[i] = ((const float4*)x)[i];
}

// ---- h0 += segment_sum(x[src], dst): one wave per edge, lane = 4 floats -----
__global__ __launch_bounds__(256) void gn_scatter_add(const float* __restrict__ x,
                                                      const int* __restrict__ ei,
                                                      float* __restrict__ h0) {
    int wave = (blockIdx.x * blockDim.x + threadIdx.x) >> 5;
    int lane = threadIdx.x & 31;
    if (wave >= N_EDGES) return;
    int src = ei[wave];
    int dst = ei[N_EDGES + wave];
    const float4 v = *(const float4*)(x + (size_t)src * D_IN + lane * 4);
    float* hp = h0 + (size_t)dst * D_IN + lane * 4;
    atomicAdd(hp + 0, v.x);
    atomicAdd(hp + 1, v.y);
    atomicAdd(hp + 2, v.z);
    atomicAdd(hp + 3, v.w);
}

// ---- WMMA GEMM: C[M,N] = leakyrelu(A[M,K] @ B[K,N] + bias) ------------------
// Block = 256 threads = 8 waves; block tile = 128 rows x 64 cols.
// Wave tile = 16 rows x 64 cols = 4x v_wmma_f32_16x16x4_f32 per K-step,
// sharing one A fragment (float2) across 4 B fragments from LDS.
// B staged in LDS in K-chunks of 128 (36 KB, pad 8 -> 2-row stride = 16 banks).
// A frag: lane l holds A[m = l&15][k0 + ((l>>4)<<1) + {0,1}]
// B frag: lane l holds B[k0 + ((l>>4)<<1) + {0,1}][n = l&15 (+16t)]
// C frag: vgpr r, lane l -> C[row0 + ((l>>4)<<3) + r][n0 + 16t + (l&15)]
template <int K, int N, bool FUSE_STATS>
__global__ __launch_bounds__(256) void gn_wmma_gemm(const float* __restrict__ A,
                                                    const float* __restrict__ B,
                                                    const float* __restrict__ bias,
                                                    float* __restrict__ C, int M,
                                                    double* __restrict__ sums,
                                                    double* __restrict__ sumsq) {
    constexpr int KC = 128;            // K-chunk staged in LDS
    __shared__ float Bs[KC][72];       // 64 cols + 8 pad (bank-conflict free)

    const int tid   = threadIdx.x;
    const int n0    = blockIdx.y * 64;
    const int wave  = tid >> 5;
    const int lane  = tid & 31;
    const int ln    = lane & 15;
    const int khalf = (lane >> 4) << 1;  // 0 or 2
    const int row0  = blockIdx.x * 128 + wave * 16;
    const int m     = row0 + ln;
    const int mc    = m < M ? m : M - 1;  // clamp: WMMA needs full EXEC
    const float* ap = A + (size_t)mc * K + khalf;

    v8f acc[4] = {};

    for (int kc = 0; kc < K; kc += KC) {
        // Stage B[kc:kc+KC, n0:n0+64] (coalesced float4 -> ds_store_b128).
        for (int idx = tid; idx < KC * 16; idx += 256) {
            int kk = idx >> 4, c4 = (idx & 15) * 4;
            float4 bv4 = *(const float4*)(B + (size_t)(kc + kk) * N + n0 + c4);
            *(float4*)&Bs[kk][c4] = bv4;
        }
        __syncthreads();

#pragma unroll 4
        for (int k = 0; k < KC; k += 4) {
            v2f a = *(const v2f*)(ap + kc + k);
#pragma unroll
            for (int t = 0; t < 4; ++t) {
                v2f b;
                b[0] = Bs[k + khalf][ln + 16 * t];
                b[1] = Bs[k + khalf + 1][ln + 16 * t];
                acc[t] = __builtin_amdgcn_wmma_f32_16x16x4_f32(
                    /*neg_a=*/false, a, /*neg_b=*/false, b,
                    /*c_mod=*/(short)0, acc[t], /*reuse_a=*/false, /*reuse_b=*/false);
            }
        }
        __syncthreads();
    }

    const int rbase = row0 + ((lane >> 4) << 3);
#pragma unroll
    for (int t = 0; t < 4; ++t) {
        const int   col = n0 + 16 * t + ln;
        const float bv  = bias[col];
        double ds = 0.0, dsq = 0.0;
#pragma unroll
        for (int r = 0; r < 8; ++r) {
            int row = rbase + r;
            float v = acc[t][r] + bv;
            v = v > 0.f ? v : v * SLOPE;
            if (row < M) {
                C[(size_t)row * N + col] = v;
                if (FUSE_STATS) { ds += (double)v; dsq += (double)v * (double)v; }
            }
        }
        if (FUSE_STATS) {  // each lane owns one column per sub-tile
            atomicAdd(&sums[col], ds);
            atomicAdd(&sumsq[col], dsq);
        }
    }
}

// ---- BN: per-column scale/shift from sums ----------------------------------
__global__ void gn_bn_finalize(const double* __restrict__ sums,
                               const double* __restrict__ sumsq,
                               const float* __restrict__ gamma,
                               const float* __restrict__ beta,
                               float* __restrict__ scale, float* __restrict__ shift) {
    int c = threadIdx.x;  // 256 threads, 1 block
    double mean = sums[c] / (double)N_NODES;
    double var  = sumsq[c] / (double)N_NODES - mean * mean;
    double inv  = 1.0 / sqrt(var + BN_EPS);
    float sc = (float)inv * gamma[c];
    scale[c] = sc;
    shift[c] = beta[c] - (float)mean * sc;
}

// ---- out = out*scale[c] + shift[c] (in place, vectorized) -------------------
__global__ void gn_bn_apply(float* __restrict__ out, const float* __restrict__ scale,
                            const float* __restrict__ shift) {
    int i = blockIdx.x * blockDim.x + threadIdx.x;
    const int total4 = N_NODES * D_OUT / 4;
    if (i >= total4) return;
    int c = (i * 4) & (D_OUT - 1);
    float4 v = ((float4*)out)[i];
    v.x = v.x * scale[c + 0] + shift[c + 0];
    v.y = v.y * scale[c + 1] + shift[c + 1];
    v.z = v.z * scale[c + 2] + shift[c + 2];
    v.w = v.w * scale[c + 3] + shift[c + 3];
    ((float4*)out)[i] = v;
}

extern "C" void kernel_launch(void* const* d_in, const int* in_sizes, int n_in,
                              void* d_out, int out_size, void* d_ws, size_t ws_size,
                              hipStream_t stream) {
    const float* x     = (const float*)d_in[0];
    const int*   ei    = (const int*)d_in[1];
    const float* W1    = (const float*)d_in[2];
    const float* b1    = (const float*)d_in[3];
    const float* W2    = (const float*)d_in[4];
    const float* b2    = (const float*)d_in[5];
    const float* gamma = (const float*)d_in[6];
    const float* beta  = (const float*)d_in[7];
    float*       out   = (float*)d_out;

    char*   ws    = (char*)d_ws;
    float*  h0    = (float*)ws;                                       // 25.6 MB
    float*  h1    = (float*)(ws + (size_t)N_NODES * D_IN * 4);        // 102.4 MB
    char*   sb    = ws + (size_t)N_NODES * (D_IN + D_HID) * 4;
    double* sums  = (double*)sb;                                      // 2 KB
    double* sumsq = (double*)(sb + 256 * 8);                          // 2 KB
    float*  scale = (float*)(sb + 2 * 256 * 8);                       // 1 KB
    float*  shift = (float*)(sb + 2 * 256 * 8 + 256 * 4);             // 1 KB

    hipMemsetAsync(sb, 0, 2 * 256 * 8, stream);  // zero BN accumulators

    const int n4 = N_NODES * D_IN / 4;
    gn_copy_x<<<(n4 + 255) / 256, 256, 0, stream>>>(x, h0, n4);
    gn_scatter_add<<<(N_EDGES + 7) / 8, 256, 0, stream>>>(x, ei, h0);

    dim3 g1((N_NODES + 127) / 128, D_HID / 64);
    gn_wmma_gemm<D_IN, D_HID, false><<<g1, 256, 0, stream>>>(
        h0, W1, b1, h1, N_NODES, nullptr, nullptr);

    dim3 g2((N_NODES + 127) / 128, D_OUT / 64);
    gn_wmma_gemm<D_HID, D_OUT, true><<<g2, 256, 0, stream>>>(
        h1, W2, b2, out, N_NODES, sums, sumsq);

    gn_bn_finalize<<<1, 256, 0, stream>>>(sums, sumsq, gamma, beta, scale, shift);

    const int t4 = N_NODES * D_OUT / 4;
    gn_bn_apply<<<(t4 + 255) / 256, 256, 0, stream>>>(out, scale, shift);
}